// GCN_61864708931601
// MI455X (gfx1250) — compile-verified
//
#include <hip/hip_runtime.h>
#include <hip/hip_bf16.h>

typedef __attribute__((ext_vector_type(2))) float v2f;
typedef __attribute__((ext_vector_type(8))) float v8f;

#define DEMB 64

// ---------------- degree / normalization ----------------

__global__ void k_deg_init(float* __restrict__ deg, int n) {
    int i = blockIdx.x * blockDim.x + threadIdx.x;
    if (i < n) deg[i] = 1.0f;               // self-loop contributes 1 to every node
}

__global__ void k_deg_count(const int* __restrict__ ei, float* __restrict__ deg, int E) {
    int e = blockIdx.x * blockDim.x + threadIdx.x;
    if (e < E) atomicAdd(&deg[ei[E + e]], 1.0f);   // col = edge_index[1][e]
}

__global__ void k_make_dis(const float* __restrict__ deg, float* __restrict__ dis, int n) {
    int i = blockIdx.x * blockDim.x + threadIdx.x;
    if (i < n) dis[i] = rsqrtf(deg[i]);     // deg >= 1 always (self loops)
}

// ---------------- dense GEMM  H[N,64] = X[N,64] @ W[64,64]  (f32 WMMA) ----------------
// One wave per 16x16 output tile; 4 waves/block cover the 64 output columns.
// V_WMMA_F32_16X16X4_F32 layouts (ISA 7.12.2):
//   A 16x4:  lanes 0-15 -> M=lane, K={0,1} in vgpr{0,1}; lanes 16-31 -> K={2,3}
//   B 4x16:  vgpr0 = rows K=0 (lanes 0-15) / K=2 (lanes 16-31); vgpr1 = K=1 / K=3
//   C 16x16: vgpr v -> row v (lanes 0-15), row v+8 (lanes 16-31), col = lane&15
__global__ void k_gemm64(const float* __restrict__ X, const float* __restrict__ W,
                         float* __restrict__ H, int n) {
    const int wid  = threadIdx.x >> 5;      // 0..3 -> column tile
    const int lane = threadIdx.x & 31;
    const int half = lane >> 4;             // 0 or 1
    const int l16  = lane & 15;
    const int r0   = blockIdx.x * 16;
    const int n0   = wid * 16;

    int row_a = r0 + l16;
    if (row_a >= n) row_a = n - 1;          // clamp (no divergence; EXEC stays full)
    const float* __restrict__ xrow = X + (size_t)row_a * DEMB;

    v8f c = {0.f, 0.f, 0.f, 0.f, 0.f, 0.f, 0.f, 0.f};
    #pragma unroll
    for (int k = 0; k < DEMB; k += 4) {
        const int kk = k + half * 2;
        v2f a, b;
        a.x = xrow[kk + 0];
        a.y = xrow[kk + 1];
        b.x = W[(kk + 0) * DEMB + n0 + l16];
        b.y = W[(kk + 1) * DEMB + n0 + l16];
        c = __builtin_amdgcn_wmma_f32_16x16x4_f32(false, a, false, b,
                                                  (short)0, c, false, false);
    }
    #pragma unroll
    for (int v = 0; v < 8; ++v) {
        const int row_c = r0 + half * 8 + v;
        if (row_c < n) H[(size_t)row_c * DEMB + n0 + l16] = c[v];
    }
}

// ---------------- aggregation ----------------

// A[i,:] = dis[i]^2 * H[i,:]   (folds the self-loop edge; also inits accumulator)
// float4-wide: 16 float4 per row -> b128 load/store
__global__ void k_self_init(const float4* __restrict__ H4, const float* __restrict__ dis,
                            float4* __restrict__ A4, int total4) {
    int idx = blockIdx.x * blockDim.x + threadIdx.x;
    if (idx < total4) {
        const int i = idx >> 4;             // node index (64/4 = 16 float4 per row)
        const float d = dis[i];
        const float s = d * d;
        float4 h = H4[idx];
        h.x *= s; h.y *= s; h.z *= s; h.w *= s;
        A4[idx] = h;
    }
}

// A[col,:] += dis[row]*dis[col] * H[row,:]
// one thread per (edge, float2): wave = one edge, b64 gather, 2 f32 atomics
__global__ void k_scatter(const int* __restrict__ ei, const float* __restrict__ dis,
                          const float* __restrict__ H, float* __restrict__ A, int E) {
    unsigned tid = blockIdx.x * 256u + threadIdx.x;
    const int e = (int)(tid >> 5);
    const int j = (int)(tid & 31u) * 2;
    if (e < E) {
        const int r = ei[e];
        const int c = ei[E + e];
        const float norm = dis[r] * dis[c];
        const float2 h = *(const float2*)(H + (size_t)r * DEMB + j);
        float* dst = A + (size_t)c * DEMB + j;
        atomicAdd(dst + 0, norm * h.x);
        atomicAdd(dst + 1, norm * h.y);
    }
}

__global__ void k_bias_act(float4* __restrict__ A4, const float4* __restrict__ b4,
                           int total4, int do_relu) {
    int idx = blockIdx.x * blockDim.x + threadIdx.x;
    if (idx < total4) {
        const float4 b = b4[idx & 15];
        float4 v = A4[idx];
        v.x += b.x; v.y += b.y; v.z += b.z; v.w += b.w;
        if (do_relu) {
            v.x = fmaxf(v.x, 0.f); v.y = fmaxf(v.y, 0.f);
            v.z = fmaxf(v.z, 0.f); v.w = fmaxf(v.w, 0.f);
        }
        A4[idx] = v;
    }
}

// ---------------- batch offsets + head ----------------

__global__ void k_zero_counts(int* __restrict__ cnt, int B) {
    int i = blockIdx.x * blockDim.x + threadIdx.x;
    if (i < B) cnt[i] = 0;
}

__global__ void k_count_batch(const int* __restrict__ batch, int* __restrict__ cnt, int n) {
    int i = blockIdx.x * blockDim.x + threadIdx.x;
    if (i < n) atomicAdd(&cnt[batch[i]], 1);
}

__global__ void k_prefix(const int* __restrict__ cnt, int* __restrict__ offs, int B) {
    if (blockIdx.x == 0 && threadIdx.x == 0) {
        int run = 0;
        for (int g = 0; g < B; ++g) { offs[g] = run; run += cnt[g]; }
    }
}

// out[g,t] = blin[t] + sum_k h2[i0,k]*Wlin[k,t] + sum_k h2[i1,k]*Wlin[64+k,t]
__global__ void k_head(const float* __restrict__ A, const int* __restrict__ offs,
                       const int* __restrict__ target, const float* __restrict__ Wlin,
                       const float* __restrict__ blin, float* __restrict__ out,
                       int B, int T) {
    int tid = blockIdx.x * blockDim.x + threadIdx.x;
    if (tid >= B * T) return;
    const int g = tid / T, t = tid % T;
    const int i0 = offs[g] + target[g * 2 + 0];
    const int i1 = offs[g] + target[g * 2 + 1];
    float acc = blin[t];
    const float* __restrict__ h0 = A + (size_t)i0 * DEMB;
    const float* __restrict__ h1 = A + (size_t)i1 * DEMB;
    #pragma unroll 8
    for (int k = 0; k < DEMB; ++k) acc += h0[k] * Wlin[k * T + t];
    #pragma unroll 8
    for (int k = 0; k < DEMB; ++k) acc += h1[k] * Wlin[(DEMB + k) * T + t];
    out[g * T + t] = acc;
}

// ---------------- launch ----------------

extern "C" void kernel_launch(void* const* d_in, const int* in_sizes, int n_in,
                              void* d_out, int out_size, void* d_ws, size_t ws_size,
                              hipStream_t stream) {
    const float* x      = (const float*)d_in[0];
    const int*   ei     = (const int*)d_in[1];   // [2,E] int32
    const int*   batch  = (const int*)d_in[2];   // [N]
    const int*   target = (const int*)d_in[3];   // [B,2]
    const float* W1     = (const float*)d_in[4];
    const float* b1     = (const float*)d_in[5];
    const float* W2     = (const float*)d_in[6];
    const float* b2     = (const float*)d_in[7];
    const float* Wlin   = (const float*)d_in[8];
    const float* blin   = (const float*)d_in[9];
    float* out = (float*)d_out;

    const int N = in_sizes[0] / DEMB;
    const int E = in_sizes[1] / 2;
    const int B = in_sizes[3] / 2;
    const int T = in_sizes[9];
    const int total  = N * DEMB;
    const int total4 = total / 4;

    // workspace layout
    float* H    = (float*)d_ws;          // [N,64]
    float* A    = H + (size_t)total;     // [N,64]
    float* deg  = A + (size_t)total;     // [N]
    float* dis  = deg + N;               // [N]
    int*   cnt  = (int*)(dis + N);       // [B]
    int*   offs = cnt + B;               // [B]

    const int TB = 256;
    const int gN    = (N + TB - 1) / TB;
    const int gE    = (E + TB - 1) / TB;
    const int gTot4 = (total4 + TB - 1) / TB;
    const unsigned gScat = (unsigned)(((size_t)E * 32 + TB - 1) / TB);
    const int gTiles = (N + 15) / 16;

    // normalization
    k_deg_init <<<gN, TB, 0, stream>>>(deg, N);
    k_deg_count<<<gE, TB, 0, stream>>>(ei, deg, E);
    k_make_dis <<<gN, TB, 0, stream>>>(deg, dis, N);

    // layer 1
    k_gemm64   <<<gTiles, 128, 0, stream>>>(x, W1, H, N);
    k_self_init<<<gTot4, TB, 0, stream>>>((const float4*)H, dis, (float4*)A, total4);
    k_scatter  <<<gScat, TB, 0, stream>>>(ei, dis, H, A, E);
    k_bias_act <<<gTot4, TB, 0, stream>>>((float4*)A, (const float4*)b1, total4, 1);

    // layer 2 (A -> GEMM -> H; then aggregate back into A, which is dead after GEMM)
    k_gemm64   <<<gTiles, 128, 0, stream>>>(A, W2, H, N);
    k_self_init<<<gTot4, TB, 0, stream>>>((const float4*)H, dis, (float4*)A, total4);
    k_scatter  <<<gScat, TB, 0, stream>>>(ei, dis, H, A, E);
    k_bias_act <<<gTot4, TB, 0, stream>>>((float4*)A, (const float4*)b2, total4, 0);

    // offsets + head
    k_zero_counts<<<(B + TB - 1) / TB, TB, 0, stream>>>(cnt, B);
    k_count_batch<<<gN, TB, 0, stream>>>(batch, cnt, N);
    k_prefix     <<<1, 1, 0, stream>>>(cnt, offs, B);
    k_head<<<(B * T + TB - 1) / TB, TB, 0, stream>>>(A, offs, target, Wlin, blin, out, B, T);
}